// Extractor_39908836114844
// MI455X (gfx1250) — compile-verified
//
#include <hip/hip_runtime.h>
#include <hip/hip_bf16.h>

// ---------------------------------------------------------------------------
// GraphSAGE x5 for MI455X (gfx1250, wave32)
//  - scatter mean-aggregation via global_atomic_add_f32 (agg array is 3.2MB,
//    L2-resident on a 192MB L2 -> atomics stay on-die)
//  - per-16-node-tile linear layers via chained V_WMMA_F32_16X16X4_F32
//    (exact fp32 math, 4 WMMAs per tile: 2 K-steps x {Wl on mean, Wr on x})
//  - L2-normalize rows with ds_swizzle_b32 xor butterflies (no address math)
//  - BN stats wave->LDS(ds_add_f32)->global atomics, applied in 2nd kernel
//  - divisions replaced by v_rcp_f32 / v_rsq_f32 (1-2 ulp)
// ---------------------------------------------------------------------------

#define NN      100000
#define HH      8
#define TILES   (NN / 16)       // 6250, exact
#define BN_EPS  1e-5f
#define L2_EPS2 1e-24f          // (1e-12)^2, compared against squared norm

typedef __attribute__((ext_vector_type(2))) float v2f;
typedef __attribute__((ext_vector_type(8))) float v8f;

// ds_swizzle_b32 group-of-32 xor mode: offset = xor_mask<<10 | and_mask(0x1F)
#define SWZ_XOR_ADD(ss, mask)                                                 \
    ss += __int_as_float(__builtin_amdgcn_ds_swizzle(                         \
        __float_as_int(ss), ((mask) << 10) | 0x1F))

// ------------------------------- utility kernels ---------------------------

__global__ void zero_f32(float* __restrict__ p, int n) {
    int i = blockIdx.x * blockDim.x + threadIdx.x;
    if (i < n) p[i] = 0.0f;
}

// pad x [N,5] -> h [N,8] with zeros in cols 5..7
__global__ void pad_input(const float* __restrict__ x, float* __restrict__ h, int total) {
    int i = blockIdx.x * blockDim.x + threadIdx.x;
    if (i >= total) return;
    int node = i >> 3;
    int col  = i & 7;
    h[i] = (col < 5) ? x[node * 5 + col] : 0.0f;
}

__global__ void count_edges(const int* __restrict__ dst, int E, float* __restrict__ cnt) {
    int e = blockIdx.x * blockDim.x + threadIdx.x;
    if (e >= E) return;
    atomicAdd(&cnt[dst[e]], 1.0f);
}

// agg[dst] += h[src]  (first F features)
__global__ void scatter_feats(const float* __restrict__ h,
                              const int* __restrict__ src,
                              const int* __restrict__ dst,
                              int E, float* __restrict__ agg, int F) {
    int e = blockIdx.x * blockDim.x + threadIdx.x;
    if (e >= E) return;
    int s = src[e];
    int d = dst[e];
    const float4* hp = (const float4*)(h + (size_t)s * HH);
    float4 lo = hp[0];
    float4 hi = hp[1];
    float v[8] = {lo.x, lo.y, lo.z, lo.w, hi.x, hi.y, hi.z, hi.w};
    float* ap = agg + (size_t)d * HH;
#pragma unroll
    for (int f = 0; f < 8; ++f)
        if (f < F) atomicAdd(&ap[f], v[f]);
}

// Pack W[H,F] row-major into padded transposed tiles WT[8][16]:
// WT[k*16+n] = W[n*F+k] for n<H && k<F else 0. 8 matrices x 128 floats.
__global__ void prep_weights(const float* __restrict__ W1l, const float* __restrict__ W1r,
                             const float* __restrict__ W2l, const float* __restrict__ W2r,
                             const float* __restrict__ W3l, const float* __restrict__ W3r,
                             const float* __restrict__ W4l, const float* __restrict__ W4r,
                             float* __restrict__ wt) {
    int t = blockIdx.x * blockDim.x + threadIdx.x;
    if (t >= 8 * 128) return;
    const float* srcs[8] = {W1l, W1r, W2l, W2r, W3l, W3r, W4l, W4r};
    int mat = t >> 7;
    int idx = t & 127;
    int k = idx >> 4;
    int n = idx & 15;
    int F = (mat < 2) ? 5 : 8;   // W1* are [8,5], rest [8,8]
    float v = (n < HH && k < F) ? srcs[mat][n * F + k] : 0.0f;
    wt[mat * 128 + idx] = v;
}

// ------------------------------ SAGE tile kernel ---------------------------
// One wave32 handles a 16-node tile: out = (agg/cnt)@Wl^T + h@Wr^T via 4
// chained V_WMMA_F32_16X16X4_F32, then L2-normalize rows, ReLU, emit pre-BN
// activations + BN partial stats.
__global__ __launch_bounds__(256) void sage_wmma(const float* __restrict__ h_in,
                                                 const float* __restrict__ agg,
                                                 const float* __restrict__ cnt,
                                                 const float* __restrict__ wtl,
                                                 const float* __restrict__ wtr,
                                                 float* __restrict__ h_pre,
                                                 float* __restrict__ stats,
                                                 int ntiles) {
    __shared__ float sh_sum[8];
    __shared__ float sh_sq[8];
    if (threadIdx.x < 8) { sh_sum[threadIdx.x] = 0.0f; sh_sq[threadIdx.x] = 0.0f; }
    __syncthreads();

    int gw   = (int)((blockIdx.x * blockDim.x + threadIdx.x) >> 5);  // global wave id
    int lane = threadIdx.x & 31;
    int n    = lane & 15;          // matrix column (B/C/D lane index)
    int half = lane >> 4;

    if (gw < ntiles) {             // wave-uniform: EXEC stays all-1 for WMMA
        int m0  = gw * 16;
        int row = m0 + n;          // A-matrix row for this lane
        float c   = fmaxf(cnt[row], 1.0f);
        float inv = __builtin_amdgcn_rcpf(c);    // v_rcp_f32, 1 ulp
        const float* aggp = agg  + (size_t)row * HH;
        const float* hp   = h_in + (size_t)row * HH;

        v8f acc = {};
#pragma unroll
        for (int s = 0; s < 4; ++s) {
            int k0 = ((s & 1) * 4) + half * 2;   // K offset for this lane/step
            v2f a, b;
            const float* bsrc;
            if (s < 2) {                          // mean @ Wl^T
                a.x = aggp[k0] * inv;
                a.y = aggp[k0 + 1] * inv;
                bsrc = wtl;
            } else {                              // x @ Wr^T
                a.x = hp[k0];
                a.y = hp[k0 + 1];
                bsrc = wtr;
            }
            b.x = bsrc[k0 * 16 + n];
            b.y = bsrc[(k0 + 1) * 16 + n];
            acc = __builtin_amdgcn_wmma_f32_16x16x4_f32(
                false, a, false, b, (short)0, acc, false, false);
        }

        // Row M = v + half*8 lives across the 16 lanes of this half-wave.
        // xor masks 1/2/4/8 stay within the half, so plain ds_swizzle xor
        // butterflies all-reduce each row's sum of squares (cols 8..15 == 0).
        float val[8];
#pragma unroll
        for (int v = 0; v < 8; ++v) {
            float d  = acc[v];
            float ss = d * d;
            SWZ_XOR_ADD(ss, 1);
            SWZ_XOR_ADD(ss, 2);
            SWZ_XOR_ADD(ss, 4);
            SWZ_XOR_ADD(ss, 8);
            // d / max(sqrt(ss), 1e-12) == d * rsqrt(max(ss, 1e-24)) (1-2 ulp)
            float o = d * __builtin_amdgcn_rsqf(fmaxf(ss, L2_EPS2));
            val[v] = fmaxf(o, 0.0f);              // ReLU
        }

        if (n < HH) {                             // single EXEC transition
            float* outp = h_pre + (size_t)(m0 + half * 8) * HH + n;
            float lsum = 0.0f, lsq = 0.0f;
#pragma unroll
            for (int v = 0; v < 8; ++v) {         // stores get imm offsets v*32B
                outp[v * HH] = val[v];
                lsum += val[v];
                lsq  += val[v] * val[v];
            }
            atomicAdd(&sh_sum[n], lsum);          // ds_add_f32
            atomicAdd(&sh_sq[n], lsq);
        }
    }

    __syncthreads();
    if (threadIdx.x < 8) {
        atomicAdd(&stats[threadIdx.x],     sh_sum[threadIdx.x]);
        atomicAdd(&stats[8 + threadIdx.x], sh_sq[threadIdx.x]);
    }
}

// BN apply: y = (x - m) * rsqrt(v + eps) * g + b, biased variance from stats.
// float4 per thread; with H=8 a 4-wide group is always cols 0-3 or 4-7.
__global__ void bn_apply(const float* __restrict__ hpre,
                         const float* __restrict__ stats,
                         const float* __restrict__ g,
                         const float* __restrict__ b,
                         float* __restrict__ out, int total4) {
    int i = blockIdx.x * blockDim.x + threadIdx.x;
    if (i >= total4) return;
    int col0 = (i * 4) & 7;      // 0 or 4
    const float invN = 1.0f / (float)NN;
    float4 v = ((const float4*)hpre)[i];
    float r[4] = {v.x, v.y, v.z, v.w};
    float o[4];
#pragma unroll
    for (int j = 0; j < 4; ++j) {
        int col = col0 + j;
        float m   = stats[col] * invN;
        float var = stats[8 + col] * invN - m * m;
        float s   = rsqrtf(var + BN_EPS);
        o[j] = (r[j] - m) * s * g[col] + b[col];
    }
    ((float4*)out)[i] = make_float4(o[0], o[1], o[2], o[3]);
}

// ---------------------------------- launcher -------------------------------

extern "C" void kernel_launch(void* const* d_in, const int* in_sizes, int n_in,
                              void* d_out, int out_size, void* d_ws, size_t ws_size,
                              hipStream_t stream) {
    const float* x   = (const float*)d_in[0];
    const int*   eic = (const int*)d_in[1];
    const int*   eid = (const int*)d_in[2];
    const float* Wp[8] = {(const float*)d_in[3], (const float*)d_in[4],
                          (const float*)d_in[5], (const float*)d_in[6],
                          (const float*)d_in[7], (const float*)d_in[8],
                          (const float*)d_in[9], (const float*)d_in[10]};
    const float* gb[8] = {(const float*)d_in[11], (const float*)d_in[12],
                          (const float*)d_in[13], (const float*)d_in[14],
                          (const float*)d_in[15], (const float*)d_in[16],
                          (const float*)d_in[17], (const float*)d_in[18]};
    const int EC = in_sizes[1] / 2;
    const int ED = in_sizes[2] / 2;

    // workspace layout (floats)
    float* ws    = (float*)d_ws;
    float* h_cur = ws;                    // N*8
    float* h_pre = h_cur + NN * HH;       // N*8
    float* agg   = h_pre + NN * HH;       // N*8
    float* cnt_c = agg + NN * HH;         // N
    float* cnt_d = cnt_c + NN;            // N
    float* stats = cnt_d + NN;            // 16 (sum[8], sumsq[8])
    float* wt    = stats + 16;            // 8 * 128

    const int T = 256;
    auto blk = [](int n, int t) { return (n + t - 1) / t; };

    // one-time per launch: counts, padded input, packed weight tiles
    zero_f32<<<blk(2 * NN, T), T, 0, stream>>>(cnt_c, 2 * NN);  // cnt_c + cnt_d
    count_edges<<<blk(EC, T), T, 0, stream>>>(eic + EC, EC, cnt_c);
    count_edges<<<blk(ED, T), T, 0, stream>>>(eid + ED, ED, cnt_d);
    pad_input<<<blk(NN * HH, T), T, 0, stream>>>(x, h_cur, NN * HH);
    prep_weights<<<4, 256, 0, stream>>>(Wp[0], Wp[1], Wp[2], Wp[3],
                                        Wp[4], Wp[5], Wp[6], Wp[7], wt);

    // layer schedule: (weight pair, edge set, bn idx, F_in)
    // conv1/eic/bn1, conv4/eic/bn2, conv2/eid/bn3, conv3/eic/bn4 (x2)
    const int  wpair[5] = {0, 3, 1, 2, 2};
    const int  bnidx[5] = {0, 1, 2, 3, 3};
    const bool useC[5]  = {true, true, false, true, true};
    const int  Fin[5]   = {5, 8, 8, 8, 8};

    const int sage_blocks = (TILES + 7) / 8;   // 8 waves per 256-thread block
    const int total4 = NN * HH / 4;

    for (int L = 0; L < 5; ++L) {
        const int*   src = useC[L] ? eic : (eid);
        const int*   dst = useC[L] ? (eic + EC) : (eid + ED);
        const int    E   = useC[L] ? EC : ED;
        const float* cnt = useC[L] ? cnt_c : cnt_d;
        const float* wtl = wt + (2 * wpair[L]) * 128;
        const float* wtr = wt + (2 * wpair[L] + 1) * 128;
        const float* g   = gb[2 * bnidx[L]];
        const float* b   = gb[2 * bnidx[L] + 1];
        float* out = (L == 4) ? (float*)d_out : h_cur;

        zero_f32<<<blk(NN * HH, T), T, 0, stream>>>(agg, NN * HH);
        zero_f32<<<1, 32, 0, stream>>>(stats, 16);
        scatter_feats<<<blk(E, T), T, 0, stream>>>(h_cur, src, dst, E, agg, Fin[L]);
        sage_wmma<<<sage_blocks, T, 0, stream>>>(h_cur, agg, cnt, wtl, wtr,
                                                 h_pre, stats, TILES);
        bn_apply<<<blk(total4, T), T, 0, stream>>>(h_pre, stats, g, b, out, total4);
    }
}